// local_emb_D_79250736545934
// MI455X (gfx1250) — compile-verified
//
#include <hip/hip_runtime.h>

typedef float v2f __attribute__((ext_vector_type(2)));
typedef float v8f __attribute__((ext_vector_type(8)));

#define HID 128
#define EPSN 1e-12f

// ---------------- Pass 1: per-row inverse L2 norm (one wave32 per row) ----------
__global__ void rownorm_kernel(const float* __restrict__ emb,
                               float* __restrict__ rnorm, int n_nodes) {
    const int wave = (blockIdx.x * blockDim.x + threadIdx.x) >> 5;
    const int lane = threadIdx.x & 31;
    if (wave >= n_nodes) return;
    // 128 floats / 32 lanes = one float4 per lane (GLOBAL_LOAD_B128)
    const float4 v = *(const float4*)(emb + (size_t)wave * HID + lane * 4);
    float ss = v.x * v.x + v.y * v.y + v.z * v.z + v.w * v.w;
#pragma unroll
    for (int off = 16; off >= 1; off >>= 1)
        ss += __shfl_xor(ss, off, 32);
    if (lane == 0)
        rnorm[wave] = 1.0f / fmaxf(sqrtf(ss), EPSN);
}

// ---------------- Pass 2: 16 edge-dots per wave via chained f32 WMMA ------------
// C[i][j] = sum_k (emb[src_i][k]*d[k]) * emb[dst_j][k]; we need only diag(C).
// A frag (16x4 f32): lane L -> row M=L&15, holds K = {kk+2*(L>>4), +1}  (8B load)
// B frag (4x16 f32): lane L -> col N=L&15, holds K = {kk+2*(L>>4), +1}  (8B load)
__global__ void edge_dot_wmma_kernel(const float* __restrict__ emb,
                                     const int* __restrict__ esrc,
                                     const int* __restrict__ edst,
                                     const float* __restrict__ dvec,
                                     const float* __restrict__ scale,
                                     const float* __restrict__ rnorm,
                                     float* __restrict__ out,
                                     int n_edges) {
    const int lane = threadIdx.x & 31;
    const int blk  = blockIdx.x * (blockDim.x >> 5) + (threadIdx.x >> 5);
    const int e0   = blk * 16;               // 16 edges per wave
    if (e0 >= n_edges) return;               // wave-uniform exit: EXEC stays all-1s

    const int m     = lane & 15;             // row/col index inside the 16-block
    const int koff  = (lane >> 4) * 2;       // K-half select
    const int eM    = min(e0 + m, n_edges - 1);   // clamp tail (branchless)
    const int si    = esrc[eM];
    const int di    = edst[eM];
    const float* __restrict__ srow = emb + (size_t)si * HID;
    const float* __restrict__ drow = emb + (size_t)di * HID;

    v8f c = {};
#pragma unroll
    for (int kk = 0; kk < HID; kk += 4) {
        v2f a        = *(const v2f*)(srow + kk + koff);
        const v2f b  = *(const v2f*)(drow + kk + koff);
        const v2f dd = *(const v2f*)(dvec + kk + koff);   // tiny, L0-resident
        a *= dd;                                          // fold diag(d) into A
        c = __builtin_amdgcn_wmma_f32_16x16x4_f32(
                /*neg_a=*/false, a, /*neg_b=*/false, b,
                /*c_mod=*/(short)0, c, /*reuse_a=*/false, /*reuse_b=*/false);
    }

    // Diagonal: lanes 0..7 -> rows 0..7 in VGPR r=lane; lanes 24..31 -> rows 8..15
    // in VGPR r=lane-24. (C layout: VGPR r, lane L holds row r+8*(L>>4), col L&15.)
    const bool valid = (lane < 8) || (lane >= 24);
    const int  r     = lane & 7;
    float diag = c[0];
#pragma unroll
    for (int j = 1; j < 8; ++j) diag = (r == j) ? c[j] : diag;

    const int e = e0 + (lane & 15);          // this lane's output edge
    if (valid && e < n_edges) {
        out[e] = diag * rnorm[si] * rnorm[di] * scale[0];
    }
}

extern "C" void kernel_launch(void* const* d_in, const int* in_sizes, int n_in,
                              void* d_out, int out_size, void* d_ws, size_t ws_size,
                              hipStream_t stream) {
    const float* emb   = (const float*)d_in[0];
    const int*   esrc  = (const int*)d_in[1];
    const int*   edst  = (const int*)d_in[2];
    const float* dvec  = (const float*)d_in[3];
    const float* scale = (const float*)d_in[4];
    const int n_nodes  = in_sizes[0] / HID;
    const int n_edges  = in_sizes[1];

    float* rnorm = (float*)d_ws;             // n_nodes floats (~400 KB)
    float* out   = (float*)d_out;

    {   // Pass 1: one wave per row, 8 waves per block
        const int threads = 256;
        const long long total = (long long)n_nodes * 32;
        const int blocks = (int)((total + threads - 1) / threads);
        rownorm_kernel<<<blocks, threads, 0, stream>>>(emb, rnorm, n_nodes);
    }
    {   // Pass 2: 16 edges per wave, 8 waves per block
        const int nblk16 = (n_edges + 15) / 16;
        const int wavesPerBlock = 8;
        const int blocks = (nblk16 + wavesPerBlock - 1) / wavesPerBlock;
        edge_dot_wmma_kernel<<<blocks, wavesPerBlock * 32, 0, stream>>>(
            emb, esrc, edst, dvec, scale, rnorm, out, n_edges);
    }
}